// VNNBlock_16003048145220
// MI455X (gfx1250) — compile-verified
//
#include <hip/hip_runtime.h>
#include <hip/hip_bf16.h>
#include <math.h>

// VNNBlock hypernetwork, MI455X / gfx1250, wave32 + V_WMMA_F32_16X16X4_F32.
//
// Key identity: with I == O == 256, the reference's torch-.repeat indexing gives
// i1 == i2 == n % 256 == o, so the hyper-weight w[s,i,o] is independent of i and
// out[s,o] = rowsum(x[s,:]) * g(s,o).  We run the hypernetwork on only
// S*O = 1024 rows instead of S*I*O = 262144 (256x less work, exact same result).
//
// Round-2 changes: branch-free main K-loop (tail K-step peeled out, loads
// select-predicated instead of exec-predicated), and pe1==pe2 reuse: one
// sin/cos pair feeds two WMMA groups (B rows k and k+64) in kernel 1.

typedef float v2f __attribute__((ext_vector_type(2)));
typedef float v8f __attribute__((ext_vector_type(8)));

#define D_MODEL 64
#define HIDDEN  128
#define I_DIM   256
#define O_DIM   256
// -ln(10000)/64 (fp32), matching numpy's div = exp(arange(0,64,2) * c)
#define PE_C   (-0.14391156831212788f)

// pe pair: (sin(o*div[j2]), cos(o*div[j2])) = (pe[o][2*j2], pe[o][2*j2+1])
__device__ __forceinline__ v2f pe_pair_j(int o, int j2 /*0..31*/) {
    float div = expf((float)(2 * j2) * PE_C);
    float ang = (float)o * div;
    float s, c;
    sincosf(ang, &s, &c);
    v2f r; r.x = s; r.y = c;
    return r;
}

__device__ __forceinline__ float half_reduce16(float v) {
    // reduce across the 16 lanes of each wave32 half (xor masks stay in-half)
    #pragma unroll
    for (int m = 8; m >= 1; m >>= 1) v += __shfl_xor(v, m, 32);
    return v;
}

// ---------------------------------------------------------------------------
// Kernel 1: h = relu(arg @ w1 + b1); g = h @ w2 + b2; out = rowsum(x) * g
// arg[row, k], row = s*256 + o:  k<128 -> pe[o][k&63] (pe1==pe2), k==128 -> x[s,o]
// One wave per 16-row tile; K padded 129 -> 132 (33 steps of K=4).
// ---------------------------------------------------------------------------
__global__ __launch_bounds__(32)
void vnn_gemm1(const float* __restrict__ x,  const float* __restrict__ w1,
               const float* __restrict__ b1, const float* __restrict__ w2,
               const float* __restrict__ b2, float* __restrict__ out_ws)
{
    const int lane = threadIdx.x;
    const int half = lane >> 4;     // 0: K+{0,1}, 1: K+{2,3}
    const int l15  = lane & 15;
    const int m0   = blockIdx.x * 16;       // row tile base (row = s*256 + o)
    const int row  = m0 + l15;              // this lane's A-matrix row
    const int o    = row & (O_DIM - 1);
    const int s    = m0 >> 8;               // uniform across the tile

    // per-lane x value for the k=128 column (loaded by ALL lanes, selected later)
    const float xval = x[s * I_DIM + o];

    // X_s = sum_i x[s, i]  (full-wave reduction; every lane ends with the total)
    float xs = 0.f;
    for (int i = lane; i < I_DIM; i += 32) xs += x[s * I_DIM + i];
    #pragma unroll
    for (int m = 16; m >= 1; m >>= 1) xs += __shfl_xor(xs, m, 32);

    v8f zero8 = {0.f, 0.f, 0.f, 0.f, 0.f, 0.f, 0.f, 0.f};
    v8f acc[8];
    #pragma unroll
    for (int nt = 0; nt < 8; ++nt) acc[nt] = zero8;

    // ---- main loop: K-steps 0..31 (k = 0..127), branch-free ----
    // A pair for step kstep equals A pair for step kstep+16 (pe1 == pe2),
    // so compute once, use against B rows kbase and kbase+64.
    for (int kstep = 0; kstep < 16; ++kstep) {
        const int kbase = kstep * 4 + half * 2;     // even, 0..62
        const v2f a = pe_pair_j(o, kbase >> 1);
        #pragma unroll
        for (int nt = 0; nt < 8; ++nt) {
            const int n = nt * 16 + l15;
            v2f b;
            b.x = w1[kbase       * HIDDEN + n];
            b.y = w1[(kbase + 1) * HIDDEN + n];
            acc[nt] = __builtin_amdgcn_wmma_f32_16x16x4_f32(
                false, a, false, b, (short)0, acc[nt], false, false);
        }
        #pragma unroll
        for (int nt = 0; nt < 8; ++nt) {
            const int n = nt * 16 + l15;
            v2f b;
            b.x = w1[(kbase + 64) * HIDDEN + n];
            b.y = w1[(kbase + 65) * HIDDEN + n];
            acc[nt] = __builtin_amdgcn_wmma_f32_16x16x4_f32(
                false, a, false, b, (short)0, acc[nt], false, false);
        }
    }

    // ---- tail K-step 32: k = 128 (x column), 129..131 pad -> zeros ----
    {
        v2f a;
        a.x = half ? 0.f : xval;                    // half 0 -> k=128, half 1 -> k=130
        a.y = 0.f;                                  // k=129 / k=131 pad
        #pragma unroll
        for (int nt = 0; nt < 8; ++nt) {
            const int n = nt * 16 + l15;
            const float w128 = w1[128 * HIDDEN + n];  // loaded by all lanes
            v2f b;
            b.x = half ? 0.f : w128;
            b.y = 0.f;
            acc[nt] = __builtin_amdgcn_wmma_f32_16x16x4_f32(
                false, a, false, b, (short)0, acc[nt], false, false);
        }
    }

    // epilogue: relu(h + b1) . w2 per row, reduce over N (16 lanes per half)
    float pr[8] = {0.f, 0.f, 0.f, 0.f, 0.f, 0.f, 0.f, 0.f};
    #pragma unroll
    for (int nt = 0; nt < 8; ++nt) {
        const int n = nt * 16 + l15;
        const float bb = b1[n], ww = w2[n];
        #pragma unroll
        for (int r = 0; r < 8; ++r) {               // C/D: VGPR r -> M = r + 8*half
            float h = acc[nt][r] + bb;
            h = h > 0.f ? h : 0.f;
            pr[r] += h * ww;
        }
    }
    const float b2v = b2[0];
    #pragma unroll
    for (int r = 0; r < 8; ++r) pr[r] = half_reduce16(pr[r]);
    if (l15 == 0) {
        #pragma unroll
        for (int r = 0; r < 8; ++r) {
            const int rr = r + 8 * half;            // row within tile
            out_ws[m0 + rr] = xs * (pr[r] + b2v);
        }
    }
}

// ---------------------------------------------------------------------------
// Kernel 2: barg = [pe[o] | out[s,o]] (65 -> pad 68); bias = relu(barg@v1+c1)@v2+c2
// final = out + bias
// ---------------------------------------------------------------------------
__global__ __launch_bounds__(32)
void vnn_gemm2(const float* __restrict__ out_ws, const float* __restrict__ v1,
               const float* __restrict__ c1,     const float* __restrict__ v2,
               const float* __restrict__ c2,     float* __restrict__ dout)
{
    const int lane = threadIdx.x;
    const int half = lane >> 4;
    const int l15  = lane & 15;
    const int m0   = blockIdx.x * 16;
    const int row  = m0 + l15;
    const int o    = row & (O_DIM - 1);

    // per-lane out value for the k=64 column (loaded by ALL lanes, selected later)
    const float outv = out_ws[row];

    v8f zero8 = {0.f, 0.f, 0.f, 0.f, 0.f, 0.f, 0.f, 0.f};
    v8f acc[8];
    #pragma unroll
    for (int nt = 0; nt < 8; ++nt) acc[nt] = zero8;

    // ---- main loop: K-steps 0..15 (k = 0..63), branch-free ----
    for (int kstep = 0; kstep < 16; ++kstep) {
        const int kbase = kstep * 4 + half * 2;     // even, 0..62
        const v2f a = pe_pair_j(o, kbase >> 1);
        #pragma unroll
        for (int nt = 0; nt < 8; ++nt) {
            const int n = nt * 16 + l15;
            v2f b;
            b.x = v1[kbase       * HIDDEN + n];
            b.y = v1[(kbase + 1) * HIDDEN + n];
            acc[nt] = __builtin_amdgcn_wmma_f32_16x16x4_f32(
                false, a, false, b, (short)0, acc[nt], false, false);
        }
    }

    // ---- tail K-step 16: k = 64 (out column), 65..67 pad -> zeros ----
    {
        v2f a;
        a.x = half ? 0.f : outv;
        a.y = 0.f;
        #pragma unroll
        for (int nt = 0; nt < 8; ++nt) {
            const int n = nt * 16 + l15;
            const float v64 = v1[64 * HIDDEN + n];  // loaded by all lanes
            v2f b;
            b.x = half ? 0.f : v64;
            b.y = 0.f;
            acc[nt] = __builtin_amdgcn_wmma_f32_16x16x4_f32(
                false, a, false, b, (short)0, acc[nt], false, false);
        }
    }

    float pr[8] = {0.f, 0.f, 0.f, 0.f, 0.f, 0.f, 0.f, 0.f};
    #pragma unroll
    for (int nt = 0; nt < 8; ++nt) {
        const int n = nt * 16 + l15;
        const float cc = c1[n], ww = v2[n];
        #pragma unroll
        for (int r = 0; r < 8; ++r) {
            float h = acc[nt][r] + cc;
            h = h > 0.f ? h : 0.f;
            pr[r] += h * ww;
        }
    }
    const float c2v = c2[0];
    #pragma unroll
    for (int r = 0; r < 8; ++r) pr[r] = half_reduce16(pr[r]);
    if (l15 == 0) {
        #pragma unroll
        for (int r = 0; r < 8; ++r) {
            const int rr = r + 8 * half;
            dout[m0 + rr] = out_ws[m0 + rr] + (pr[r] + c2v);
        }
    }
}

extern "C" void kernel_launch(void* const* d_in, const int* in_sizes, int n_in,
                              void* d_out, int out_size, void* d_ws, size_t ws_size,
                              hipStream_t stream) {
    const float* x  = (const float*)d_in[0];   // [4, 256]
    const float* w1 = (const float*)d_in[1];   // [129, 128]
    const float* b1 = (const float*)d_in[2];   // [128]
    const float* w2 = (const float*)d_in[3];   // [128] (of [128,1])
    const float* b2 = (const float*)d_in[4];   // [1]
    const float* v1 = (const float*)d_in[5];   // [65, 128]
    const float* c1 = (const float*)d_in[6];   // [128]
    const float* v2 = (const float*)d_in[7];   // [128] (of [128,1])
    const float* c2 = (const float*)d_in[8];   // [1]
    (void)in_sizes; (void)n_in; (void)out_size; (void)ws_size;

    float* out_ws = (float*)d_ws;              // 1024 floats: pre-bias out[s,o]
    float* out    = (float*)d_out;             // 1024 floats: final [4, 256]

    // 1024 rows / 16 rows-per-wave = 64 tiles; 1 wave (32 threads) per tile.
    vnn_gemm1<<<64, 32, 0, stream>>>(x, w1, b1, w2, b2, out_ws);
    vnn_gemm2<<<64, 32, 0, stream>>>(out_ws, v1, c1, v2, c2, out);
}